// Encoder_17600775979896
// MI455X (gfx1250) — compile-verified
//
#include <hip/hip_runtime.h>

// ---------------- constants from the reference ----------------
#define NN      500000L     // nodes
#define NE      8000000L    // edges
#define NG      4096L       // graphs
#define D_IN    32
#define D_H     16
#define D_OUT   32

typedef __attribute__((ext_vector_type(2))) float v2f;
typedef __attribute__((ext_vector_type(8))) float v8f;

// ---------------- small utility kernels ----------------
__global__ void k_fill(float* p, float v, long n) {
    long i = (long)blockIdx.x * blockDim.x + threadIdx.x;
    if (i < n) p[i] = v;
}

// deg[dst] += 1 per edge
__global__ void k_deg_accum(const int* __restrict__ dst, float* __restrict__ deg, long ne) {
    long e = (long)blockIdx.x * blockDim.x + threadIdx.x;
    if (e < ne) atomicAdd(&deg[dst[e]], 1.0f);
}

// deg -> rsqrt(deg) in place (becomes "dis")
__global__ void k_rsqrt(float* p, long n) {
    long i = (long)blockIdx.x * blockDim.x + threadIdx.x;
    if (i < n) p[i] = rsqrtf(p[i]);
}

// ---------------- edge aggregation: agg[dst] += dis[src]*dis[dst]*xw[src] ----------------
// F consecutive lanes handle one edge -> coalesced row gather + coalesced atomics.
template <int F>
__global__ void k_edge_agg(const int* __restrict__ src, const int* __restrict__ dst,
                           const float* __restrict__ dis, const float* __restrict__ xw,
                           float* __restrict__ agg, long ne) {
    long t = (long)blockIdx.x * blockDim.x + threadIdx.x;
    long e = t / F;
    int  f = (int)(t % F);
    if (e >= ne) return;
    int s = src[e];
    int d = dst[e];
    float norm = dis[s] * dis[d];
    atomicAdd(&agg[(long)d * F + f], norm * xw[(long)s * F + f]);
}

// h = relu(agg + dis^2 * xw + bias)   (h may alias agg)
template <int F>
__global__ void k_selfloop_relu(const float* __restrict__ agg, const float* __restrict__ xw,
                                const float* __restrict__ dis, const float* __restrict__ bias,
                                float* __restrict__ h, long nn) {
    long t = (long)blockIdx.x * blockDim.x + threadIdx.x;
    long n = t / F;
    int  f = (int)(t % F);
    if (n >= nn) return;
    float ds = dis[n];
    float v = agg[n * F + f] + (ds * ds) * xw[n * F + f] + bias[f];
    h[n * F + f] = fmaxf(v, 0.0f);
}

// ---------------- fp32 WMMA GEMM: C[M,N] = A[M,K] @ B[K,N] (+bias, relu) ----------------
// One wave (32 lanes) per 16x16 output tile; blockDim.x == 32 so EXEC is all-1s.
// Layouts per CDNA5 ISA 7.12.2 (32-bit A 16x4, B/C row-striped across lanes).
__global__ void k_wmma_gemm_f32(const float* __restrict__ A, const float* __restrict__ B,
                                float* __restrict__ C, int M, int N, int K,
                                const float* __restrict__ bias, int relu) {
    int tm   = blockIdx.x;            // M tile
    int tn   = blockIdx.y;            // N tile
    int lane = threadIdx.x;
    int half = lane >> 4;             // 0: lanes 0-15, 1: lanes 16-31
    int l    = lane & 15;

    long rowA = (long)tm * 16 + l;    // A: M = l in both halves
    int  col  = tn * 16 + l;          // B/C: N = l in both halves

    v8f c = {};
    for (int k = 0; k < K; k += 4) {
        // A 16x4: vgpr0 holds K = k+2*half, vgpr1 holds K = k+2*half+1
        const float* ap = A + rowA * K + k + 2 * half;
        v2f a; a.x = ap[0]; a.y = ap[1];
        // B 4x16: vgpr0 row K = k+2*half, vgpr1 row K = k+2*half+1
        int kb = k + 2 * half;
        v2f b; b.x = B[(long)kb * N + col];
               b.y = B[(long)(kb + 1) * N + col];
        c = __builtin_amdgcn_wmma_f32_16x16x4_f32(false, a, false, b,
                                                  (short)0, c, false, false);
    }
    float bb = bias ? bias[col] : 0.0f;
    #pragma unroll
    for (int r = 0; r < 8; ++r) {
        long row = (long)tm * 16 + r + 8 * half;   // C: vgpr r -> M = r (+8 upper half)
        float v = c[r] + bb;
        if (relu) v = fmaxf(v, 0.0f);
        C[row * N + col] = v;
    }
}

// ---------------- pooling ----------------
__global__ void k_pool_accum(const float* __restrict__ h, const int* __restrict__ batch,
                             float* __restrict__ sums, float* __restrict__ counts, long nn) {
    long t = (long)blockIdx.x * blockDim.x + threadIdx.x;
    long n = t / D_OUT;
    int  f = (int)(t % D_OUT);
    if (n >= nn) return;
    int g = batch[n];
    atomicAdd(&sums[(long)g * D_OUT + f], h[n * D_OUT + f]);
    if (f == 0) atomicAdd(&counts[g], 1.0f);
}

__global__ void k_pool_div(float* __restrict__ sums, const float* __restrict__ counts, long ng) {
    long t = (long)blockIdx.x * blockDim.x + threadIdx.x;
    long g = t / D_OUT;
    int  f = (int)(t % D_OUT);
    if (g >= ng) return;
    sums[g * D_OUT + f] /= fmaxf(counts[g], 1.0f);
}

// ---------------- launcher ----------------
static inline long cdiv(long a, long b) { return (a + b - 1) / b; }

extern "C" void kernel_launch(void* const* d_in, const int* in_sizes, int n_in,
                              void* d_out, int out_size, void* d_ws, size_t ws_size,
                              hipStream_t stream) {
    const float* x     = (const float*)d_in[0];
    const int*   ei    = (const int*)d_in[1];     // [2, NE]: src row then dst row
    /* d_in[2] edge_attr unused by reference */
    const int*   batch = (const int*)d_in[3];
    /* d_in[4] num_graphs == 4096 (compile-time constant) */
    const float* W1 = (const float*)d_in[5];
    const float* b1 = (const float*)d_in[6];
    const float* W2 = (const float*)d_in[7];
    const float* b2 = (const float*)d_in[8];
    const float* Wl = (const float*)d_in[9];
    const float* bl = (const float*)d_in[10];
    float* out = (float*)d_out;

    const int* src = ei;
    const int* dst = ei + NE;

    // workspace layout (floats)
    float* ws = (float*)d_ws;
    float* deg    = ws;                      // NN      (becomes dis = rsqrt(deg))
    float* xw1    = deg    + NN;             // NN*16
    float* h1     = xw1    + NN * D_H;       // NN*16   (agg1, then h1 in place)
    float* xw2    = h1     + NN * D_H;       // NN*32
    float* h2     = xw2    + NN * D_OUT;     // NN*32   (agg2, then h2 in place)
    float* sums   = h2     + NN * D_OUT;     // NG*32
    float* counts = sums   + NG * D_OUT;     // NG

    const int B = 256;

    // ---- degrees: deg = 1 (self loop) + indegree; then dis = rsqrt(deg)
    k_fill<<<cdiv(NN, B), B, 0, stream>>>(deg, 1.0f, NN);
    k_deg_accum<<<cdiv(NE, B), B, 0, stream>>>(dst, deg, NE);
    k_rsqrt<<<cdiv(NN, B), B, 0, stream>>>(deg, NN);

    // ---- layer 1: xw1 = x @ W1   (500000x32 @ 32x16, WMMA f32)
    {
        dim3 grid((unsigned)(NN / 16), D_H / 16);
        k_wmma_gemm_f32<<<grid, 32, 0, stream>>>(x, W1, xw1, (int)NN, D_H, D_IN, nullptr, 0);
    }
    hipMemsetAsync(h1, 0, sizeof(float) * NN * D_H, stream);
    k_edge_agg<D_H><<<cdiv(NE * D_H, B), B, 0, stream>>>(src, dst, deg, xw1, h1, NE);
    k_selfloop_relu<D_H><<<cdiv(NN * D_H, B), B, 0, stream>>>(h1, xw1, deg, b1, h1, NN);

    // ---- layer 2: xw2 = h1 @ W2  (500000x16 @ 16x32, WMMA f32)
    {
        dim3 grid((unsigned)(NN / 16), D_OUT / 16);
        k_wmma_gemm_f32<<<grid, 32, 0, stream>>>(h1, W2, xw2, (int)NN, D_OUT, D_H, nullptr, 0);
    }
    hipMemsetAsync(h2, 0, sizeof(float) * NN * D_OUT, stream);
    k_edge_agg<D_OUT><<<cdiv(NE * D_OUT, B), B, 0, stream>>>(src, dst, deg, xw2, h2, NE);
    k_selfloop_relu<D_OUT><<<cdiv(NN * D_OUT, B), B, 0, stream>>>(h2, xw2, deg, b2, h2, NN);

    // ---- global mean pool
    hipMemsetAsync(sums, 0, sizeof(float) * (NG * D_OUT + NG), stream);
    k_pool_accum<<<cdiv(NN * D_OUT, B), B, 0, stream>>>(h2, batch, sums, counts, NN);
    k_pool_div<<<cdiv(NG * D_OUT, B), B, 0, stream>>>(sums, counts, NG);

    // ---- final: out = relu(pooled @ Wl + bl)  (4096x32 @ 32x32, WMMA f32)
    {
        dim3 grid((unsigned)(NG / 16), D_OUT / 16);
        k_wmma_gemm_f32<<<grid, 32, 0, stream>>>(sums, Wl, out, (int)NG, D_OUT, D_OUT, bl, 1);
    }
}